// RoIAlignAvg_31464930410633
// MI455X (gfx1250) — compile-verified
//
#include <hip/hip_runtime.h>

// RoIAlignAvg on gfx1250 via separable interpolation expressed as two small
// fp32 WMMA GEMM chains per RoI:
//   out[c] (7x7) = U (7xR) * F_box[c] (RxX) * V (Xx7)
// with channels mapped to the WMMA M dimension (16 ch / tile).
// V_WMMA_F32_16X16X4_F32, wave32.

typedef __attribute__((ext_vector_type(2))) float v2f;
typedef __attribute__((ext_vector_type(8))) float v8f;

#define POOL   7      // final output grid
#define PHW    8      // aligned grid (pool input)
#define BOX    20     // padded bounding-box rows/cols (>=18 needed, mult of 4)
#define CPW    64     // channels per workgroup (C=256 -> 4 quarters)
#define C_TOT  256
#define FH     64
#define FW     64
#define SCALEF 0.0625f

__global__ __launch_bounds__(256) void roialign_avg_wmma(
    const float* __restrict__ feat,   // [4,256,64,64]
    const float* __restrict__ rois,   // [N,5]
    float* __restrict__ out,          // [N,256,7,7]
    int n_rois)
{
  __shared__ float sU[BOX][16];          // U^T: [r_local][i]   (cols 7..15 stay 0)
  __shared__ float sV[BOX][16];          // V  : [x_local][j]   (cols 7..15 stay 0)
  __shared__ float sT[CPW][POOL][BOX];   // stage-1 result T[c_local][i][x_local]

  const int tid  = threadIdx.x;
  const int wave = tid >> 5;
  const int lane = tid & 31;
  const int half = lane >> 4;      // 0: lanes 0-15, 1: lanes 16-31
  const int lan  = lane & 15;

  const int n  = blockIdx.x >> 2;  // roi index
  const int cq = blockIdx.x & 3;   // channel quarter
  if (n >= n_rois) return;         // wave-uniform; EXEC stays all-ones

  // ---- RoI parameters (computed redundantly per thread) ----
  const float* rp = rois + (size_t)n * 5;
  const int   bidx = (int)rp[0];
  const float x1 = rp[1] * SCALEF, y1 = rp[2] * SCALEF;
  const float x2 = rp[3] * SCALEF, y2 = rp[4] * SCALEF;
  const float roiw = fmaxf(x2 - x1, 1.0f);
  const float roih = fmaxf(y2 - y1, 1.0f);
  const float bw = roiw / (float)PHW;
  const float bh = roih / (float)PHW;

  // min clipped integer sample coords (sample coords are monotone in sy/sx)
  const float yy0 = y1 + 0.25f * bh;
  const float xx0 = x1 + 0.25f * bw;
  const int rmin = (int)floorf(fminf(fmaxf(yy0, 0.0f), (float)(FH - 1)));
  const int cmin = (int)floorf(fminf(fmaxf(xx0, 0.0f), (float)(FW - 1)));

  // ---- zero the tiny weight matrices ----
  for (int i = tid; i < BOX * 16; i += blockDim.x) {
    (&sU[0][0])[i] = 0.0f;
    (&sV[0][0])[i] = 0.0f;
  }
  __syncthreads();

  // ---- build U (threads 0..6 -> column i) and V (threads 8..14 -> column j).
  // Each sample weight carries 1/4 (mean over 2x2 samples) * 1/4 (avgpool),
  // split as 0.25 per axis. Validity factors separably into U and V.
  if (tid < POOL) {
    const int i = tid;
    for (int d = 0; d < 4; ++d) {
      const int sy = 2 * i + d;                       // sample row in 0..15
      const float off = (sy & 1) ? 0.75f : 0.25f;
      const float yy = y1 + ((float)(sy >> 1) + off) * bh;
      const bool valid = (yy >= -1.0f) && (yy <= (float)FH);
      const float yc = fminf(fmaxf(yy, 0.0f), (float)(FH - 1));
      const int y0 = (int)floorf(yc);
      const float ly = yc - (float)y0;
      const int y1i = min(y0 + 1, FH - 1);
      if (valid) {
        const int r0 = min(max(y0  - rmin, 0), BOX - 1);
        const int r1 = min(max(y1i - rmin, 0), BOX - 1);
        sU[r0][i] += 0.25f * (1.0f - ly);
        sU[r1][i] += 0.25f * ly;
      }
    }
  } else if (tid >= 8 && tid < 8 + POOL) {
    const int j = tid - 8;
    for (int d = 0; d < 4; ++d) {
      const int sx = 2 * j + d;
      const float off = (sx & 1) ? 0.75f : 0.25f;
      const float xx = x1 + ((float)(sx >> 1) + off) * bw;
      const bool valid = (xx >= -1.0f) && (xx <= (float)FW);
      const float xc = fminf(fmaxf(xx, 0.0f), (float)(FW - 1));
      const int x0 = (int)floorf(xc);
      const float lx = xc - (float)x0;
      const int x1i = min(x0 + 1, FW - 1);
      if (valid) {
        const int c0 = min(max(x0  - cmin, 0), BOX - 1);
        const int c1 = min(max(x1i - cmin, 0), BOX - 1);
        sV[c0][j] += 0.25f * (1.0f - lx);
        sV[c1][j] += 0.25f * lx;
      }
    }
  }
  __syncthreads();

  // ---- Stage 1: per box column x, T[c, i, x] = sum_r F[c, rmin+r, x] * U[i, r]
  // A (16x4 f32, MxK): lane = channel (M), VGPR {0,1} x half = K
  // B (4x16 f32, KxN): lane = i (N),       VGPR {0,1} x half = K
  const size_t cbase = ((size_t)bidx * C_TOT + (size_t)cq * CPW) * (FH * FW);
  for (int p = wave; p < 4 * BOX; p += 8) {          // wave-uniform loop
    const int ctile = p / BOX;                        // 0..3
    const int x     = p % BOX;                        // 0..19
    const int xg    = min(cmin + x, FW - 1);
    const int cl    = ctile * 16 + lan;               // local channel 0..63
    const float* fc = feat + cbase + (size_t)cl * (FH * FW) + xg;

    v8f acc = {};
#pragma unroll
    for (int k0 = 0; k0 < BOX; k0 += 4) {
      const int rr = k0 + half * 2;
      const int r0 = min(rmin + rr,     FH - 1);      // clamped; weight 0 anyway
      const int r1 = min(rmin + rr + 1, FH - 1);
      v2f a, b;
      a.x = fc[(size_t)r0 * FW];
      a.y = fc[(size_t)r1 * FW];
      b.x = sU[rr][lan];
      b.y = sU[rr + 1][lan];
      acc = __builtin_amdgcn_wmma_f32_16x16x4_f32(
          false, a, false, b, (short)0, acc, false, false);
    }
    // D (16x16 f32): VGPR q -> M = q + half*8 (channel), N = lan (i)
    if (lan < POOL) {
      const int i  = lan;
      const int cs = ctile * 16 + half * 8;
#pragma unroll
      for (int q = 0; q < 8; ++q)
        sT[cs + q][i][x] = acc[q];
    }
  }
  __syncthreads();

  // ---- Stage 2: per output row i, out[c, i, j] = sum_x T[c, i, x] * V[x, j]
  for (int p = wave; p < 4 * POOL; p += 8) {          // wave-uniform loop
    const int ctile = p / POOL;                       // 0..3
    const int i     = p % POOL;                       // 0..6
    const int cl    = ctile * 16 + lan;

    v8f acc = {};
#pragma unroll
    for (int k0 = 0; k0 < BOX; k0 += 4) {
      const int xx = k0 + half * 2;
      v2f a, b;
      a.x = sT[cl][i][xx];
      a.y = sT[cl][i][xx + 1];
      b.x = sV[xx][lan];
      b.y = sV[xx + 1][lan];
      acc = __builtin_amdgcn_wmma_f32_16x16x4_f32(
          false, a, false, b, (short)0, acc, false, false);
    }
    if (lan < POOL) {
      const int j = lan;
#pragma unroll
      for (int q = 0; q < 8; ++q) {
        const int cg = cq * CPW + ctile * 16 + half * 8 + q;
        out[(((size_t)n * C_TOT + cg) * POOL + i) * POOL + j] = acc[q];
      }
    }
  }
}

extern "C" void kernel_launch(void* const* d_in, const int* in_sizes, int n_in,
                              void* d_out, int out_size, void* d_ws, size_t ws_size,
                              hipStream_t stream) {
  const float* feat = (const float*)d_in[0];   // [4,256,64,64] fp32
  const float* rois = (const float*)d_in[1];   // [N,5] fp32
  float* out = (float*)d_out;                  // [N,256,7,7] fp32
  const int n_rois = in_sizes[1] / 5;

  dim3 grid((unsigned)(n_rois * 4));           // (roi, channel-quarter)
  dim3 block(256);                             // 8 wave32 waves
  roialign_avg_wmma<<<grid, block, 0, stream>>>(feat, rois, out, n_rois);
}